// FilterDetections_51419348467826
// MI455X (gfx1250) — compile-verified
//
#include <hip/hip_runtime.h>
#include <stdint.h>

// ---------------- problem constants ----------------
#define BB 4
#define NN 100000
#define CC 80
#define KK 256          // PRE_NMS_K
#define MAXDET 300
#define CAP 2048        // candidate cap per (b,c) and per image
#define BC (BB*CC)      // 320
#define M_IMG (NN*CC)   // 8,000,000 elements per image
#define SCORE_THR 0.05f
#define NMS_THR 0.5f

// tiling for streaming scans
#define TILE 4096                  // floats per async tile (16KB)
#define FULL_TILES 1953            // 1953*4096 = 7,999,488
#define TAIL_START (FULL_TILES*TILE)
#define TAIL_N 512                 // 8,000,000 - 7,999,488
#define TPB 8                      // tiles per block
#define BLKS_PER_IMG 245           // ceil(1954/8)

// ---------------- workspace layout ----------------
// u32-indexed region (zeroed each launch)
#define OFF_H1     0               // 320*64
#define OFF_H2     20480
#define OFF_H3     40960
#define OFF_CUTA   61440           // 320
#define OFF_ABOVEA 61760
#define OFF_CUTB   62080
#define OFF_ABOVEB 62400
#define OFF_T      62720
#define OFF_CNT    63040
#define ZERO_N     63360
// byte offsets
#define BYTE_CAND    253440ull                         // 320*2048*8 = 5,242,880
#define BYTE_CLSKEYS (BYTE_CAND + 5242880ull)          // 320*256*8  = 655,360
#define BYTE_CLSBOX  (BYTE_CLSKEYS + 655360ull)        // 320*256*16 = 1,310,720

typedef float v2f __attribute__((ext_vector_type(2)));
typedef float v8f __attribute__((ext_vector_type(8)));

// ---------------- CDNA5 async global->LDS helpers ----------------
__device__ __forceinline__ unsigned lds_off(const void* p) {
  return (unsigned)(uintptr_t)p;   // low 32 bits of flat shared addr = LDS byte offset
}
__device__ __forceinline__ void async_load_b128(unsigned lds_byte, const float* g) {
  asm volatile("global_load_async_to_lds_b128 %0, %1, off"
               :: "v"(lds_byte), "v"((unsigned long long)(uintptr_t)g)
               : "memory");
}
__device__ __forceinline__ void wait_async4() { asm volatile("s_wait_asynccnt 0x4" ::: "memory"); }
__device__ __forceinline__ void wait_async0() { asm volatile("s_wait_asynccnt 0x0" ::: "memory"); }

// ---------------- zero kernel ----------------
__global__ void fd_zero_kernel(unsigned* wsU, int n) {
  int i = blockIdx.x * blockDim.x + threadIdx.x;
  if (i < n) wsU[i] = 0u;
}

// ---------------- streaming scan (hist levels 1-3, collect=4) ----------------
template <int MODE>
__device__ __forceinline__ void process_elem(unsigned e, float val,
                                             const unsigned* cutA_l, const unsigned* cutB_l,
                                             const unsigned* T_l, unsigned* histL,
                                             unsigned* gcnt, unsigned long long* cand,
                                             unsigned bBase) {
  unsigned bits = __float_as_uint(val);
  unsigned c = e % (unsigned)CC;
  if (MODE == 1) {
    unsigned bin = bits >> 24; if (bin > 63u) bin = 63u;
    atomicAdd(&histL[c * 64u + bin], 1u);
  } else if (MODE == 2) {
    if ((bits >> 24) == cutA_l[c])
      atomicAdd(&histL[c * 64u + ((bits >> 18) & 63u)], 1u);
  } else if (MODE == 3) {
    if ((bits >> 24) == cutA_l[c] && ((bits >> 18) & 63u) == cutB_l[c])
      atomicAdd(&histL[c * 64u + ((bits >> 12) & 63u)], 1u);
  } else {
    if (bits >= T_l[c]) {
      unsigned pos = atomicAdd(&gcnt[bBase + c], 1u);
      if (pos < (unsigned)CAP)
        cand[(size_t)(bBase + c) * CAP + pos] =
            (((unsigned long long)bits) << 32) | (unsigned long long)(0xFFFFFFFFu - e / (unsigned)CC);
    }
  }
}

template <int MODE>
__global__ void fd_scan_kernel(const float* __restrict__ cls, unsigned* wsU,
                               unsigned long long* cand) {
  __shared__ float buf[2][TILE];
  __shared__ unsigned histL[CC * 64];
  __shared__ unsigned cutA_l[CC], cutB_l[CC], T_l[CC];

  const int tid = threadIdx.x;                  // 256 threads
  const int b = blockIdx.x / BLKS_PER_IMG;
  const int tb = blockIdx.x % BLKS_PER_IMG;
  const float* img = cls + (size_t)b * M_IMG;

  if (MODE != 4) for (int i = tid; i < CC * 64; i += 256) histL[i] = 0u;
  if (tid < CC) {
    cutA_l[tid] = wsU[OFF_CUTA + b * CC + tid];
    cutB_l[tid] = wsU[OFF_CUTB + b * CC + tid];
    T_l[tid]    = wsU[OFF_T    + b * CC + tid];
  }
  __syncthreads();

  unsigned* gcnt = wsU + OFF_CNT;
  const unsigned bBase = (unsigned)(b * CC);

  int t0 = tb * TPB;
  int nt = FULL_TILES - t0; if (nt > TPB) nt = TPB; if (nt < 0) nt = 0;

  // prefetch tile 0 (4 x b128 per thread = 16 floats)
  if (nt > 0) {
    const float* g = img + (size_t)t0 * TILE + tid * 16;
    unsigned l = lds_off(&buf[0][0]) + tid * 64;
#pragma unroll
    for (int q = 0; q < 4; ++q) async_load_b128(l + q * 16, g + q * 4);
  }

  for (int i = 0; i < nt; ++i) {
    if (i + 1 < nt) {
      const float* g = img + (size_t)(t0 + i + 1) * TILE + tid * 16;
      unsigned l = lds_off(&buf[(i + 1) & 1][0]) + tid * 64;
#pragma unroll
      for (int q = 0; q < 4; ++q) async_load_b128(l + q * 16, g + q * 4);
      wait_async4();   // in-order: previous tile's 4 loads complete
    } else {
      wait_async0();
    }
    unsigned eBase = (unsigned)(t0 + i) * TILE + (unsigned)tid * 16u;
    const float4* lb4 = (const float4*)(&buf[i & 1][tid * 16]);
#pragma unroll
    for (int q = 0; q < 4; ++q) {
      float4 v4 = lb4[q];
      process_elem<MODE>(eBase + q * 4 + 0, v4.x, cutA_l, cutB_l, T_l, histL, gcnt, cand, bBase);
      process_elem<MODE>(eBase + q * 4 + 1, v4.y, cutA_l, cutB_l, T_l, histL, gcnt, cand, bBase);
      process_elem<MODE>(eBase + q * 4 + 2, v4.z, cutA_l, cutB_l, T_l, histL, gcnt, cand, bBase);
      process_elem<MODE>(eBase + q * 4 + 3, v4.w, cutA_l, cutB_l, T_l, histL, gcnt, cand, bBase);
    }
  }

  // tail (512 floats), plain loads
  if (tb == BLKS_PER_IMG - 1) {
    for (int idx = tid; idx < TAIL_N; idx += 256) {
      unsigned e = (unsigned)TAIL_START + (unsigned)idx;
      process_elem<MODE>(e, img[e], cutA_l, cutB_l, T_l, histL, gcnt, cand, bBase);
    }
  }

  if (MODE != 4) {
    __syncthreads();
    const unsigned off = (MODE == 1) ? OFF_H1 : (MODE == 2) ? OFF_H2 : OFF_H3;
    for (int i = tid; i < CC * 64; i += 256) {
      unsigned v = histL[i];
      if (v) atomicAdd(&wsU[off + (unsigned)(b * CC) * 64u + (unsigned)i], v);
    }
  }
}

// ---------------- per-(b,c) cutoff finder, 3 phases ----------------
__global__ void fd_cutoff_kernel(unsigned* wsU, int phase) {
  int bc = blockIdx.x * blockDim.x + threadIdx.x;
  if (bc >= BC) return;
  if (phase == 0) {
    const unsigned* h = wsU + OFF_H1 + bc * 64;
    unsigned above = 0, cut = 0;
    for (int bin = 63; bin >= 0; --bin) {
      unsigned v = h[bin];
      if (above + v >= (unsigned)KK) { cut = (unsigned)bin; break; }
      above += v;
    }
    wsU[OFF_CUTA + bc] = cut; wsU[OFF_ABOVEA + bc] = above;
  } else if (phase == 1) {
    const unsigned* h = wsU + OFF_H2 + bc * 64;
    unsigned target = (unsigned)KK - wsU[OFF_ABOVEA + bc];
    unsigned above = 0, cut = 0;
    for (int bin = 63; bin >= 0; --bin) {
      unsigned v = h[bin];
      if (above + v >= target) { cut = (unsigned)bin; break; }
      above += v;
    }
    wsU[OFF_CUTB + bc] = cut;
    wsU[OFF_ABOVEB + bc] = wsU[OFF_ABOVEA + bc] + above;
  } else {
    const unsigned* h = wsU + OFF_H3 + bc * 64;
    unsigned target = (unsigned)KK - wsU[OFF_ABOVEB + bc];
    unsigned above = 0, cut = 0;
    for (int bin = 63; bin >= 0; --bin) {
      unsigned v = h[bin];
      if (above + v >= target) { cut = (unsigned)bin; break; }
      above += v;
    }
    wsU[OFF_T + bc] = (wsU[OFF_CUTA + bc] << 24) | (wsU[OFF_CUTB + bc] << 18) | (cut << 12);
  }
}

// ---------------- in-LDS bitonic sort, 2048 u64, descending, 256 threads ----------------
__device__ __forceinline__ void bitonic2048_desc(unsigned long long* keys, int tid) {
  for (unsigned ksz = 2; ksz <= 2048; ksz <<= 1) {
    for (unsigned j = ksz >> 1; j > 0; j >>= 1) {
      __syncthreads();
      for (unsigned i = (unsigned)tid; i < 2048u; i += 256u) {
        unsigned l = i ^ j;
        if (l > i) {
          unsigned long long a = keys[i], c = keys[l];
          bool up = ((i & ksz) == 0);            // descending overall
          if (up ? (a < c) : (a > c)) { keys[i] = c; keys[l] = a; }
        }
      }
    }
  }
  __syncthreads();
}

// ---------------- per-(b,c) sort + NMS (uses WMMA for area_i + area_j) ----------------
__global__ void fd_nms_kernel(const float* __restrict__ boxes, unsigned* wsU,
                              const unsigned long long* __restrict__ cand,
                              unsigned long long* __restrict__ clskeys,
                              float* __restrict__ clsbox) {
  const int bc = blockIdx.x;           // 0..319
  const int b = bc / CC, c = bc % CC;
  const int tid = threadIdx.x;         // 256

  __shared__ unsigned long long keys[CAP];
  __shared__ float sx1[KK], sy1[KK], sx2[KK], sy2[KK], sarea[KK];
  __shared__ unsigned maskL[KK * 8];
  __shared__ unsigned char validL[KK];
  __shared__ unsigned keepW[8];

  unsigned n = wsU[OFF_CNT + bc]; if (n > (unsigned)CAP) n = CAP;
  const unsigned long long* my = cand + (size_t)bc * CAP;
  for (int i = tid; i < CAP; i += 256) keys[i] = (i < (int)n) ? my[i] : 0ull;
  bitonic2048_desc(keys, tid);

  // slot tid of top-256
  unsigned long long key = keys[tid];
  unsigned sBits = (unsigned)(key >> 32);
  float score = __uint_as_float(sBits);
  unsigned idx = 0xFFFFFFFFu - (unsigned)key;
  float x1 = 0.f, y1 = 0.f, x2 = 0.f, y2 = 0.f;
  if (key != 0ull && idx < (unsigned)NN) {
    const float* bp = boxes + ((size_t)b * NN + idx) * 4;
    x1 = bp[0]; y1 = bp[1]; x2 = bp[2]; y2 = bp[3];
  }
  sx1[tid] = x1; sy1[tid] = y1; sx2[tid] = x2; sy2[tid] = y2;
  sarea[tid] = (x2 - x1) * (y2 - y1);
  validL[tid] = (score > SCORE_THR) ? 1 : 0;
  for (int i = tid; i < KK * 8; i += 256) maskL[i] = 0u;
  __syncthreads();

  // pairwise overlap bitmask; areasum tile via V_WMMA_F32_16X16X4_F32
  const int wv = tid >> 5, lane = tid & 31;
  for (int t = wv; t < 256; t += 8) {      // 16x16 tiles of the 256x256 matrix
    int ti = t >> 4, tj = t & 15;
    if (tj > ti) continue;                 // only need col < row
    int al = ti * 16 + (lane & 15);
    int bl = tj * 16 + (lane & 15);
    v2f A, Bf;
    A.x = (lane < 16) ? sarea[al] : 0.0f;  // A[m][0] = area_i
    A.y = (lane < 16) ? 1.0f : 0.0f;       // A[m][1] = 1
    Bf.x = (lane < 16) ? 1.0f : 0.0f;      // B[0][n] = 1
    Bf.y = (lane < 16) ? sarea[bl] : 0.0f; // B[1][n] = area_j
    v8f Cz = {0.f, 0.f, 0.f, 0.f, 0.f, 0.f, 0.f, 0.f};
    v8f D = __builtin_amdgcn_wmma_f32_16x16x4_f32(false, A, false, Bf, (short)0, Cz,
                                                  false, false);  // D = area_i + area_j
#pragma unroll
    for (int v = 0; v < 8; ++v) {
      int row = ti * 16 + v + ((lane >= 16) ? 8 : 0);
      int col = tj * 16 + (lane & 15);
      if (col < row) {
        float lt0 = fmaxf(sx1[row], sx1[col]);
        float lt1 = fmaxf(sy1[row], sy1[col]);
        float rb0 = fminf(sx2[row], sx2[col]);
        float rb1 = fminf(sy2[row], sy2[col]);
        float w = rb0 - lt0; if (w < 0.f) w = 0.f;
        float h = rb1 - lt1; if (h < 0.f) h = 0.f;
        float inter = w * h;
        float iou = inter / (D[v] - inter + 1e-9f);
        if (iou > NMS_THR)
          atomicOr(&maskL[row * 8 + (col >> 5)], 1u << (col & 31));
      }
    }
  }
  __syncthreads();

  // sequential greedy keep (matches reference fori_loop exactly)
  if (tid == 0) {
    unsigned kw[8] = {0, 0, 0, 0, 0, 0, 0, 0};
    for (int i = 0; i < KK; ++i) {
      unsigned sup = 0;
      const unsigned* mi = &maskL[i * 8];
#pragma unroll
      for (int w = 0; w < 8; ++w) sup |= mi[w] & kw[w];
      if (validL[i] && sup == 0u) kw[i >> 5] |= (1u << (i & 31));
    }
#pragma unroll
    for (int w = 0; w < 8; ++w) keepW[w] = kw[w];
  }
  __syncthreads();

  bool kept = ((keepW[tid >> 5] >> (tid & 31)) & 1u) != 0u;
  unsigned flat = (unsigned)(c * KK + tid);
  clskeys[(size_t)bc * KK + tid] =
      kept ? ((((unsigned long long)sBits) << 32) | (unsigned long long)(0xFFFFFFFFu - flat)) : 0ull;
  float* cb = clsbox + ((size_t)bc * KK + tid) * 4;
  cb[0] = x1; cb[1] = y1; cb[2] = x2; cb[3] = y2;
}

// ---------------- per-image top-300 + output ----------------
__global__ void fd_final_kernel(unsigned* wsU,
                                const unsigned long long* __restrict__ clskeys,
                                const float* __restrict__ clsbox,
                                float* __restrict__ out) {
  const int b = blockIdx.x;      // 0..3
  const int tid = threadIdx.x;   // 256
  const unsigned long long* ck = clskeys + (size_t)b * CC * KK;  // 20480 keys

  __shared__ unsigned hist[64];
  __shared__ unsigned sv[8];     // 0:cut1 1:above1 2:cut2 3:above2tot 5:T 6:selAll 7:cnt
  __shared__ unsigned long long cand2[CAP];

  // level 1
  if (tid < 64) hist[tid] = 0u;
  __syncthreads();
  for (int i = tid; i < CC * KK; i += 256) {
    unsigned long long k = ck[i];
    if (k) {
      unsigned bits = (unsigned)(k >> 32);
      unsigned bin = bits >> 24; if (bin > 63u) bin = 63u;
      atomicAdd(&hist[bin], 1u);
    }
  }
  __syncthreads();
  if (tid == 0) {
    unsigned above = 0; int cut = -1;
    for (int bin = 63; bin >= 0; --bin) {
      unsigned v = hist[bin];
      if (above + v >= (unsigned)MAXDET) { cut = bin; break; }
      above += v;
    }
    if (cut < 0) { sv[6] = 1u; sv[5] = 0u; }
    else { sv[6] = 0u; sv[0] = (unsigned)cut; sv[1] = above; }
  }
  __syncthreads();

  if (sv[6] == 0u) {
    unsigned cut1 = sv[0];
    // level 2
    if (tid < 64) hist[tid] = 0u;
    __syncthreads();
    for (int i = tid; i < CC * KK; i += 256) {
      unsigned long long k = ck[i];
      if (k) {
        unsigned bits = (unsigned)(k >> 32);
        if ((bits >> 24) == cut1) atomicAdd(&hist[(bits >> 18) & 63u], 1u);
      }
    }
    __syncthreads();
    if (tid == 0) {
      unsigned target = (unsigned)MAXDET - sv[1];
      unsigned above = 0, cut = 0;
      for (int bin = 63; bin >= 0; --bin) {
        unsigned v = hist[bin];
        if (above + v >= target) { cut = (unsigned)bin; break; }
        above += v;
      }
      sv[2] = cut; sv[3] = sv[1] + above;
    }
    __syncthreads();
    unsigned cut2 = sv[2];
    // level 3
    if (tid < 64) hist[tid] = 0u;
    __syncthreads();
    for (int i = tid; i < CC * KK; i += 256) {
      unsigned long long k = ck[i];
      if (k) {
        unsigned bits = (unsigned)(k >> 32);
        if ((bits >> 24) == cut1 && ((bits >> 18) & 63u) == cut2)
          atomicAdd(&hist[(bits >> 12) & 63u], 1u);
      }
    }
    __syncthreads();
    if (tid == 0) {
      unsigned target = (unsigned)MAXDET - sv[3];
      unsigned above = 0, cut = 0;
      for (int bin = 63; bin >= 0; --bin) {
        unsigned v = hist[bin];
        if (above + v >= target) { cut = (unsigned)bin; break; }
        above += v;
      }
      sv[5] = (cut1 << 24) | (cut2 << 18) | (cut << 12);
    }
    __syncthreads();
  }

  // collect
  if (tid == 0) sv[7] = 0u;
  __syncthreads();
  unsigned T = sv[5];
  for (int i = tid; i < CC * KK; i += 256) {
    unsigned long long k = ck[i];
    if (k && ((unsigned)(k >> 32) >= T)) {
      unsigned pos = atomicAdd(&sv[7], 1u);
      if (pos < (unsigned)CAP) cand2[pos] = k;
    }
  }
  __syncthreads();
  unsigned scnt = sv[7]; if (scnt > (unsigned)CAP) scnt = CAP;
  for (int i = tid; i < CAP; i += 256) if ((unsigned)i >= scnt) cand2[i] = 0ull;
  __syncthreads();
  bitonic2048_desc(cand2, tid);

  // write top-300
  for (int t = tid; t < MAXDET; t += 256) {
    unsigned long long k = cand2[t];
    float score = 0.f; int label = -1;
    float bx0 = 0.f, bx1 = 0.f, bx2 = 0.f, bx3 = 0.f;
    if (k) {
      unsigned bits = (unsigned)(k >> 32);
      score = __uint_as_float(bits);
      unsigned flat = 0xFFFFFFFFu - (unsigned)k;       // c*256 + rank
      label = (int)(flat >> 8);
      const float* cb = clsbox + ((size_t)b * CC * KK + flat) * 4;
      bx0 = cb[0]; bx1 = cb[1]; bx2 = cb[2]; bx3 = cb[3];
    }
    float* ob = out + ((size_t)b * MAXDET + t) * 4;
    ob[0] = bx0; ob[1] = bx1; ob[2] = bx2; ob[3] = bx3;
    out[(size_t)BB * MAXDET * 4 + b * MAXDET + t] = score;
    ((int*)out)[(size_t)BB * MAXDET * 5 + b * MAXDET + t] = label;
  }
}

// ---------------- launch ----------------
extern "C" void kernel_launch(void* const* d_in, const int* in_sizes, int n_in,
                              void* d_out, int out_size, void* d_ws, size_t ws_size,
                              hipStream_t stream) {
  (void)in_sizes; (void)n_in; (void)out_size; (void)ws_size;
  const float* boxes = (const float*)d_in[0];
  const float* cls = (const float*)d_in[1];
  float* out = (float*)d_out;
  unsigned* wsU = (unsigned*)d_ws;
  unsigned long long* cand = (unsigned long long*)((char*)d_ws + BYTE_CAND);
  unsigned long long* clskeys = (unsigned long long*)((char*)d_ws + BYTE_CLSKEYS);
  float* clsbox = (float*)((char*)d_ws + BYTE_CLSBOX);

  fd_zero_kernel<<<(ZERO_N + 255) / 256, 256, 0, stream>>>(wsU, ZERO_N);

  fd_scan_kernel<1><<<BB * BLKS_PER_IMG, 256, 0, stream>>>(cls, wsU, cand);
  fd_cutoff_kernel<<<(BC + 63) / 64, 64, 0, stream>>>(wsU, 0);
  fd_scan_kernel<2><<<BB * BLKS_PER_IMG, 256, 0, stream>>>(cls, wsU, cand);
  fd_cutoff_kernel<<<(BC + 63) / 64, 64, 0, stream>>>(wsU, 1);
  fd_scan_kernel<3><<<BB * BLKS_PER_IMG, 256, 0, stream>>>(cls, wsU, cand);
  fd_cutoff_kernel<<<(BC + 63) / 64, 64, 0, stream>>>(wsU, 2);
  fd_scan_kernel<4><<<BB * BLKS_PER_IMG, 256, 0, stream>>>(cls, wsU, cand);

  fd_nms_kernel<<<BC, 256, 0, stream>>>(boxes, wsU, cand, clskeys, clsbox);
  fd_final_kernel<<<BB, 256, 0, stream>>>(wsU, clskeys, clsbox, out);
}